// VNL_Loss_66133906424171
// MI455X (gfx1250) — compile-verified
//
#include <hip/hip_runtime.h>
#include <math.h>

#define HH 480
#define WW 640

typedef unsigned int u32;

__constant__ float kFX = 519.0f;
__constant__ float kFY = 519.0f;
__constant__ float kU0 = 320.0f;
__constant__ float kV0 = 240.0f;

// ---------------------------------------------------------------------------
// CDNA5 async data mover helpers (gfx1250): global -> LDS async copy, tracked
// by ASYNCcnt (CDNA5 ISA). The generic pointer for a
// __shared__ object is {SHARED_BASE_hi32, lds_offset_lo32}, so truncating to
// 32 bits yields the LDS byte address the VDST operand expects.
// ---------------------------------------------------------------------------
__device__ __forceinline__ void async_g2lds_b32(const void* gptr, void* lptr) {
    unsigned lds = (unsigned)(size_t)lptr;
    unsigned long long ga = (unsigned long long)(size_t)gptr;
    asm volatile("global_load_async_to_lds_b32 %0, %1, off"
                 :: "v"(lds), "v"(ga) : "memory");
}
__device__ __forceinline__ void wait_async0() {
    asm volatile("s_wait_asynccnt 0" ::: "memory");
}

// ---------------------------------------------------------------------------
// Kernel 0: zero histograms / control block / partials each call
// (harness poisons d_ws once and never re-poisons between graph replays).
// ---------------------------------------------------------------------------
__global__ void vnl_init(u32* __restrict__ hist1, u32* __restrict__ hist2,
                         u32* __restrict__ ctrl, float* __restrict__ psum,
                         u32* __restrict__ pcnt) {
    int i = blockIdx.x * blockDim.x + threadIdx.x;   // 256*256 = 65536 threads
    hist1[i] = 0u;
    hist2[i] = 0u;
    if (i < 16) ctrl[i] = 0u;
    if (i < 512) { psum[i] = 0.0f; pcnt[i] = 0u; }
}

// ---------------------------------------------------------------------------
// Kernel 1: per-(b,g) geometry + fused hi-16 histogram of MASKED keys only
// (sentinel keys are never histogrammed -> no same-address atomic storm).
// One block = 256 groups x all B batches; indices staged once via async->LDS.
// ---------------------------------------------------------------------------
__global__ void __launch_bounds__(256)
vnl_compute(const float* __restrict__ gt, const float* __restrict__ pr,
            const int* __restrict__ p1x, const int* __restrict__ p1y,
            const int* __restrict__ p2x, const int* __restrict__ p2y,
            const int* __restrict__ p3x, const int* __restrict__ p3y,
            u32* __restrict__ keys, u32* __restrict__ ctrl,
            u32* __restrict__ hist1, int B_, int G_) {
    __shared__ int sidx[6][256];
    __shared__ u32 blkc;
    const int t = threadIdx.x;
    int g = blockIdx.x * 256 + t;
    const bool active = (g < G_);
    const int gc = active ? g : (G_ - 1);

    if (t == 0) blkc = 0u;

    // CDNA5 async global->LDS staging of the six index streams
    async_g2lds_b32(&p1x[gc], &sidx[0][t]);
    async_g2lds_b32(&p1y[gc], &sidx[1][t]);
    async_g2lds_b32(&p2x[gc], &sidx[2][t]);
    async_g2lds_b32(&p2y[gc], &sidx[3][t]);
    async_g2lds_b32(&p3x[gc], &sidx[4][t]);
    async_g2lds_b32(&p3y[gc], &sidx[5][t]);
    wait_async0();
    __syncthreads();

    const int X1 = sidx[0][t], Y1 = sidx[1][t];
    const int X2 = sidx[2][t], Y2 = sidx[3][t];
    const int X3 = sidx[4][t], Y3 = sidx[5][t];
    const int o1 = Y1 * WW + X1, o2 = Y2 * WW + X2, o3 = Y3 * WW + X3;

    // warm the gather cachelines for the first batch (global_prefetch_b8)
    __builtin_prefetch(gt + o1, 0, 3);
    __builtin_prefetch(gt + o2, 0, 3);
    __builtin_prefetch(gt + o3, 0, 3);
    __builtin_prefetch(pr + o1, 0, 3);
    __builtin_prefetch(pr + o2, 0, 3);
    __builtin_prefetch(pr + o3, 0, 3);

    const float u1 = (float)X1 - kU0, v1 = (float)Y1 - kV0;
    const float u2 = (float)X2 - kU0, v2 = (float)Y2 - kV0;
    const float u3 = (float)X3 - kU0, v3 = (float)Y3 - kV0;

    u32 mcount = 0u;
    for (int b = 0; b < B_; ++b) {
        const float* gi = gt + (size_t)b * (HH * WW);
        const float* pi = pr + (size_t)b * (HH * WW);
        const float g1 = gi[o1], g2 = gi[o2], g3 = gi[o3];
        const float q1 = pi[o1], q2 = pi[o2], q3 = pi[o3];

        // ---- GT 3D points A,B,C -------------------------------------------
        float ax = u1 * fabsf(g1) / kFX, ay = v1 * fabsf(g1) / kFY, az = g1;
        float bx = u2 * fabsf(g2) / kFX, by = v2 * fabsf(g2) / kFY, bz = g2;
        float cx = u3 * fabsf(g3) / kFX, cy = v3 * fabsf(g3) / kFY, cz = g3;

        // difference vectors D0=p2-p1, D1=p3-p1, D2=p3-p2
        float D0x = bx - ax, D0y = by - ay, D0z = bz - az;
        float D1x = cx - ax, D1y = cy - ay, D1z = cz - az;
        float D2x = cx - bx, D2y = cy - by, D2z = cz - bz;

        // 3x3 Gram (einsum) + norms -> cos mask
        float e00 = D0x * D0x + D0y * D0y + D0z * D0z;
        float e11 = D1x * D1x + D1y * D1y + D1z * D1z;
        float e22 = D2x * D2x + D2y * D2y + D2z * D2z;
        float e01 = D0x * D1x + D0y * D1y + D0z * D1z;
        float e02 = D0x * D2x + D0y * D2y + D0z * D2z;
        float e12 = D1x * D2x + D1y * D2y + D1z * D2z;
        float n0 = sqrtf(e00), n1 = sqrtf(e11), n2 = sqrtf(e22);
        int cc = 0;
        cc += (fabsf(e00 / (n0 * n0 + 1e-8f)) > 0.867f) ? 1 : 0;
        cc += (fabsf(e11 / (n1 * n1 + 1e-8f)) > 0.867f) ? 1 : 0;
        cc += (fabsf(e22 / (n2 * n2 + 1e-8f)) > 0.867f) ? 1 : 0;
        cc += (fabsf(e01 / (n0 * n1 + 1e-8f)) > 0.867f) ? 2 : 0;
        cc += (fabsf(e02 / (n0 * n2 + 1e-8f)) > 0.867f) ? 2 : 0;
        cc += (fabsf(e12 / (n1 * n2 + 1e-8f)) > 0.867f) ? 2 : 0;
        const bool mask_cos = (cc > 3);
        const bool mx = (fabsf(D0x) < 0.01f) || (fabsf(D1x) < 0.01f) || (fabsf(D2x) < 0.01f);
        const bool my = (fabsf(D0y) < 0.01f) || (fabsf(D1y) < 0.01f) || (fabsf(D2y) < 0.01f);
        const bool mz = (fabsf(D0z) < 0.01f) || (fabsf(D1z) < 0.01f) || (fabsf(D2z) < 0.01f);
        const bool m = !((mx && my && mz) || mask_cos);

        // ---- pred 3D points, P[coord][point], with reference's zmask quirk:
        // zmask indexed by *point* but broadcast over the *coord* axis.
        float P[3][3];
        P[0][0] = u1 * fabsf(q1) / kFX; P[1][0] = v1 * fabsf(q1) / kFY; P[2][0] = q1;
        P[0][1] = u2 * fabsf(q2) / kFX; P[1][1] = v2 * fabsf(q2) / kFY; P[2][1] = q2;
        P[0][2] = u3 * fabsf(q3) / kFX; P[1][2] = v3 * fabsf(q3) / kFY; P[2][2] = q3;
        const bool z0 = (P[2][0] == 0.0f), z1 = (P[2][1] == 0.0f), z2 = (P[2][2] == 0.0f);
        if (z0) { P[0][0] = P[0][1] = P[0][2] = 1e-4f; }
        if (z1) { P[1][0] = P[1][1] = P[1][2] = 1e-4f; }
        if (z2) { P[2][0] = P[2][1] = P[2][2] = 1e-4f; }

        const float q12x = P[0][1] - P[0][0], q12y = P[1][1] - P[1][0], q12z = P[2][1] - P[2][0];
        const float q13x = P[0][2] - P[0][0], q13y = P[1][2] - P[1][0], q13z = P[2][2] - P[2][0];

        // normals (gt uses unmodified points: D0 = p12, D1 = p13)
        const float gnx = D0y * D1z - D0z * D1y;
        const float gny = D0z * D1x - D0x * D1z;
        const float gnz = D0x * D1y - D0y * D1x;
        const float dnx = q12y * q13z - q12z * q13y;
        const float dny = q12z * q13x - q12x * q13z;
        const float dnz = q12x * q13y - q12y * q13x;
        float gl = sqrtf(gnx * gnx + gny * gny + gnz * gnz); if (gl == 0.0f) gl = 0.01f;
        float dl = sqrtf(dnx * dnx + dny * dny + dnz * dnz); if (dl == 0.0f) dl = 0.01f;
        const float L = fabsf(gnx / gl - dnx / dl) + fabsf(gny / gl - dny / dl) +
                        fabsf(gnz / gl - dnz / dl);

        if (active) {
            const u32 key = m ? __float_as_uint(L) : 0xFFFFFFFFu;
            keys[(size_t)b * G_ + g] = key;
            if (m) {
                ++mcount;
                atomicAdd(&hist1[key >> 16], 1u);  // masked keys only: bins well spread
            }
        }
    }
    atomicAdd(&blkc, mcount);
    __syncthreads();
    if (t == 0) atomicAdd(&ctrl[0], blkc);
}

// ---------------------------------------------------------------------------
// Kernel 3: low-16 histogram of keys inside the selected hi-bin
// (sentinel keys have hi=0xFFFF != b1, so they never match).
// ---------------------------------------------------------------------------
__global__ void vnl_hist_lo(const u32* __restrict__ keys, const u32* __restrict__ ctrl,
                            u32* __restrict__ hist, int NT) {
    const u32 b1 = ctrl[1];
    int stride = gridDim.x * blockDim.x;
    for (int i = blockIdx.x * blockDim.x + threadIdx.x; i < NT; i += stride) {
        const u32 k = keys[i];
        if ((k >> 16) == b1) atomicAdd(&hist[k & 0xFFFFu], 1u);
    }
}

// ---------------------------------------------------------------------------
// Kernel 2: coarse radix-select. k = drop+1 = (n/4)+1; find hi-16 bin + rank.
// ---------------------------------------------------------------------------
__global__ void vnl_coarse(const u32* __restrict__ hist, u32* __restrict__ ctrl) {
    __shared__ u32 csum[256];
    const int t = threadIdx.x;
    u32 s = 0;
    for (int i = 0; i < 256; ++i) s += hist[t * 256 + i];
    csum[t] = s;
    __syncthreads();
    if (t == 0) {
        const u32 n = ctrl[0];
        if (n == 0u) {
            ctrl[3] = 0xFFFFFFFEu;  // nothing exceeds this except the sentinel
            ctrl[4] = 0u;
        } else {
            ctrl[4] = 1u;
            const u32 drop = n / 4u;
            const u32 k = drop + 1u;     // rank of smallest KEPT value
            u32 cum = 0; int ch = 0;
            for (; ch < 256; ++ch) { if (cum + csum[ch] >= k) break; cum += csum[ch]; }
            if (ch > 255) ch = 255;
            u32 rem = k - cum, c2 = 0;
            u32 bin = (u32)ch * 256u;
            for (int i = 0; i < 256; ++i) {
                const u32 h = hist[ch * 256 + i];
                if (c2 + h >= rem) { bin = (u32)ch * 256u + (u32)i; rem -= c2; break; }
                c2 += h;
            }
            ctrl[1] = bin;   // hi 16 bits of threshold
            ctrl[2] = rem;   // rank within that bin
        }
    }
}

// ---------------------------------------------------------------------------
// Kernel 4: fine radix-select -> exact 32-bit threshold key T in ctrl[3]
// ---------------------------------------------------------------------------
__global__ void vnl_fine(const u32* __restrict__ hist, u32* __restrict__ ctrl) {
    __shared__ u32 csum[256];
    const int t = threadIdx.x;
    u32 s = 0;
    for (int i = 0; i < 256; ++i) s += hist[t * 256 + i];
    csum[t] = s;
    __syncthreads();
    if (t == 0 && ctrl[4] != 0u) {
        const u32 k = ctrl[2];
        u32 cum = 0; int ch = 0;
        for (; ch < 256; ++ch) { if (cum + csum[ch] >= k) break; cum += csum[ch]; }
        if (ch > 255) ch = 255;
        u32 rem = k - cum, c2 = 0;
        u32 lo = (u32)ch * 256u;
        for (int i = 0; i < 256; ++i) {
            const u32 h = hist[ch * 256 + i];
            if (c2 + h >= rem) { lo = (u32)ch * 256u + (u32)i; break; }
            c2 += h;
        }
        ctrl[3] = (ctrl[1] << 16) | lo;
    }
}

// ---------------------------------------------------------------------------
// Kernel 5: fixed-order partial sums of masked values strictly above T
// (deterministic: fixed grid, fixed per-thread stride order, fixed tree).
// ---------------------------------------------------------------------------
__global__ void __launch_bounds__(256)
vnl_sum(const u32* __restrict__ keys, const u32* __restrict__ ctrl,
        float* __restrict__ psum, u32* __restrict__ pcnt, int NT) {
    const u32 T = ctrl[3];
    const int t = threadIdx.x;
    float s = 0.0f; u32 c = 0;
    const int stride = gridDim.x * blockDim.x;
    for (int i = blockIdx.x * blockDim.x + t; i < NT; i += stride) {
        const u32 k = keys[i];
        if (k != 0xFFFFFFFFu && k > T) { s += __uint_as_float(k); ++c; }
    }
    __shared__ float ss[256];
    __shared__ u32 sc[256];
    ss[t] = s; sc[t] = c;
    __syncthreads();
    for (int o = 128; o > 0; o >>= 1) {
        if (t < o) { ss[t] += ss[t + o]; sc[t] += sc[t + o]; }
        __syncthreads();
    }
    if (t == 0) { psum[blockIdx.x] = ss[0]; pcnt[blockIdx.x] = sc[0]; }
}

// ---------------------------------------------------------------------------
// Kernel 6: final reduce + trimmed-mean formula.
// sum(top kept) = sum(>T) + (kept - count(>T)) * T ;  kept = n - n/4.
// ---------------------------------------------------------------------------
__global__ void vnl_final(const u32* __restrict__ ctrl, const float* __restrict__ psum,
                          const u32* __restrict__ pcnt, float* __restrict__ out) {
    __shared__ float ss[512];
    __shared__ u32 sc[512];
    const int t = threadIdx.x;
    ss[t] = psum[t]; sc[t] = pcnt[t];
    __syncthreads();
    for (int o = 256; o > 0; o >>= 1) {
        if (t < o) { ss[t] += ss[t + o]; sc[t] += sc[t + o]; }
        __syncthreads();
    }
    if (t == 0) {
        const u32 n = ctrl[0];
        float r = 0.0f;
        if (n > 0u) {
            const u32 drop = n / 4u;
            const u32 kept = n - drop;
            const float T = __uint_as_float(ctrl[3]);
            const u32 ties = kept - sc[0];
            r = (ss[0] + (float)ties * T) / (float)kept;
        }
        out[0] = r;
    }
}

// ---------------------------------------------------------------------------
extern "C" void kernel_launch(void* const* d_in, const int* in_sizes, int n_in,
                              void* d_out, int out_size, void* d_ws, size_t ws_size,
                              hipStream_t stream) {
    const float* gt = (const float*)d_in[0];
    const float* pr = (const float*)d_in[1];
    const int* p1x = (const int*)d_in[2];
    const int* p1y = (const int*)d_in[3];
    const int* p2x = (const int*)d_in[4];
    const int* p2y = (const int*)d_in[5];
    const int* p3x = (const int*)d_in[6];
    const int* p3y = (const int*)d_in[7];

    const int G_ = in_sizes[2];
    const int B_ = in_sizes[0] / (HH * WW);
    const int NT = B_ * G_;

    // workspace carve-up (≈12.4 MB total)
    unsigned char* w = (unsigned char*)d_ws;
    u32* keys = (u32*)w;            w += (size_t)NT * sizeof(u32);
    w = (unsigned char*)(((size_t)w + 255) & ~(size_t)255);
    u32* hist1 = (u32*)w;           w += 65536u * sizeof(u32);
    u32* hist2 = (u32*)w;           w += 65536u * sizeof(u32);
    u32* ctrl = (u32*)w;            w += 64u * sizeof(u32);
    float* psum = (float*)w;        w += 512u * sizeof(float);
    u32* pcnt = (u32*)w;

    vnl_init<<<256, 256, 0, stream>>>(hist1, hist2, ctrl, psum, pcnt);
    vnl_compute<<<(G_ + 255) / 256, 256, 0, stream>>>(gt, pr, p1x, p1y, p2x, p2y,
                                                      p3x, p3y, keys, ctrl, hist1,
                                                      B_, G_);
    vnl_coarse<<<1, 256, 0, stream>>>(hist1, ctrl);
    vnl_hist_lo<<<2048, 256, 0, stream>>>(keys, ctrl, hist2, NT);
    vnl_fine<<<1, 256, 0, stream>>>(hist2, ctrl);
    vnl_sum<<<512, 256, 0, stream>>>(keys, ctrl, psum, pcnt, NT);
    vnl_final<<<1, 512, 0, stream>>>(ctrl, psum, pcnt, (float*)d_out);
}